// NonbondedTorchForce_75419625717905
// MI455X (gfx1250) — compile-verified
//
#include <hip/hip_runtime.h>

#define CUTOFF2f 0.81f
#define PREFACF  138.93544539709032f

typedef float v2f __attribute__((ext_vector_type(2)));
typedef float v8f __attribute__((ext_vector_type(8)));

// ---------------------------------------------------------------------------
// Full 32-lane wave sum using V_WMMA_F32_16X16X4_F32.
// A (16x4 f32): VGPR0 = per-lane partial, VGPR1 = 0
//   -> A[m][0] = partial[m] (lanes 0-15), A[m][2] = partial[m+16] (lanes 16-31)
// B (4x16) = all ones  ->  D[m][n] = partial[m] + partial[m+16]
// D layout: lane l<16 holds rows 0..7 in d[0..7]; lanes 16..31 hold rows 8..15.
// Sum of the 8 D components + shfl_xor(16) = total over all 32 lanes.
// Requires EXEC == all ones (call sites are fully convergent).
// ---------------------------------------------------------------------------
__device__ __forceinline__ float wave_sum32_wmma(float x) {
  v2f a; a[0] = x;    a[1] = 0.0f;
  v2f b; b[0] = 1.0f; b[1] = 1.0f;
  v8f c = {};
  v8f d = __builtin_amdgcn_wmma_f32_16x16x4_f32(false, a, false, b,
                                                (short)0, c, false, false);
  float s = ((d[0] + d[1]) + (d[2] + d[3])) + ((d[4] + d[5]) + (d[6] + d[7]));
  s += __shfl_xor(s, 16, 32);
  return s;  // total replicated to every lane
}

__device__ __forceinline__ float block_sum256(float x) {
  __shared__ float smem[8];
  const int lane = threadIdx.x & 31;
  const int wid  = threadIdx.x >> 5;
  float w = wave_sum32_wmma(x);
  if (lane == 0) smem[wid] = w;
  __syncthreads();
  float v = (lane < 8) ? smem[lane] : 0.0f;
  return wave_sum32_wmma(v);  // every wave computes the same block total
}

// ---------------------------------------------------------------------------
// Per-pair energy (Coulomb + LJ with minimum-image PBC), matching reference.
// ---------------------------------------------------------------------------
template <bool PACKED>
__device__ __forceinline__ float pair_energy(
    int i, int j,
    const float4* __restrict__ pq, const float2* __restrict__ se,
    const float* __restrict__ coords, const float* __restrict__ charges,
    const float* __restrict__ sigma,  const float* __restrict__ eps,
    float Lx, float Ly, float Lz, float iLx, float iLy, float iLz)
{
  float xi, yi, zi, qi, sgi, epi, xj, yj, zj, qj, sgj, epj;
  if (PACKED) {
    float4 A = pq[i], B = pq[j];
    float2 U = se[i], V = se[j];
    xi = A.x; yi = A.y; zi = A.z; qi = A.w;
    xj = B.x; yj = B.y; zj = B.z; qj = B.w;
    sgi = U.x; epi = U.y; sgj = V.x; epj = V.y;
  } else {
    xi = coords[3*i]; yi = coords[3*i+1]; zi = coords[3*i+2];
    xj = coords[3*j]; yj = coords[3*j+1]; zj = coords[3*j+2];
    qi = charges[i];  qj = charges[j];
    sgi = sigma[i];   sgj = sigma[j];
    epi = eps[i];     epj = eps[j];
  }
  float dx = xi - xj, dy = yi - yj, dz = zi - zj;
  dx -= Lx * rintf(dx * iLx);            // jnp.round == rint (half-to-even)
  dy -= Ly * rintf(dy * iLy);
  dz -= Lz * rintf(dz * iLz);
  float r2 = dx*dx + dy*dy + dz*dz;

  bool valid = ((i / 3) != (j / 3)) && (r2 < CUTOFF2f);
  float safe_r2 = valid ? r2 : 1.0f;

  float inv_r  = rsqrtf(safe_r2);
  float inv_r2 = inv_r * inv_r;
  float e_coul = PREFACF * qi * qj * inv_r;
  float sg  = 0.5f * (sgi + sgj);
  float ep  = sqrtf(epi * epj);
  float sr2 = sg * sg * inv_r2;
  float sr6 = sr2 * sr2 * sr2;
  float e_lj = 4.0f * ep * (sr6 * sr6 - sr6);
  return valid ? (e_coul + e_lj) : 0.0f;
}

// ---------------------------------------------------------------------------
// Prologue: pack particle data AoS so each gather is b128 + b64 (L2 resident).
// ---------------------------------------------------------------------------
__global__ void __launch_bounds__(256) pack_params_kernel(
    const float* __restrict__ coords, const float* __restrict__ charges,
    const float* __restrict__ sigma,  const float* __restrict__ eps,
    float4* __restrict__ pq, float2* __restrict__ se, int n)
{
  int i = blockIdx.x * blockDim.x + threadIdx.x;
  if (i < n) {
    pq[i] = make_float4(coords[3*i], coords[3*i+1], coords[3*i+2], charges[i]);
    se[i] = make_float2(sigma[i], eps[i]);
  }
}

// ---------------------------------------------------------------------------
// Main pass: stream pairs as int4 (2 pairs/load) with prefetch, gather params,
// accumulate per-thread, WMMA block reduce, write per-block partial.
// ---------------------------------------------------------------------------
template <bool PACKED>
__global__ void __launch_bounds__(256) nb_energy_kernel(
    const int4* __restrict__ pairs4, int nPair4,
    const int*  __restrict__ pairsFlat, int nPairs,
    const float4* __restrict__ pq, const float2* __restrict__ se,
    const float* __restrict__ coords, const float* __restrict__ charges,
    const float* __restrict__ sigma,  const float* __restrict__ eps,
    const float* __restrict__ box,
    float* __restrict__ partials)
{
  const float Lx = box[0], Ly = box[4], Lz = box[8];
  const float iLx = 1.0f / Lx, iLy = 1.0f / Ly, iLz = 1.0f / Lz;

  const int tid    = blockIdx.x * blockDim.x + threadIdx.x;
  const int stride = gridDim.x * blockDim.x;

  float acc = 0.0f;
  for (int k = tid; k < nPair4; k += stride) {
    int4 p = pairs4[k];
    __builtin_prefetch((const void*)(pairs4 + k + 4 * stride), 0, 0);  // global_prefetch_b8
    acc += pair_energy<PACKED>(p.x, p.y, pq, se, coords, charges, sigma, eps,
                               Lx, Ly, Lz, iLx, iLy, iLz);
    acc += pair_energy<PACKED>(p.z, p.w, pq, se, coords, charges, sigma, eps,
                               Lx, Ly, Lz, iLx, iLy, iLz);
  }
  // Odd tail pair (if any) handled by one thread; branch reconverges before WMMA.
  if (blockIdx.x == 0 && threadIdx.x == 0) {
    for (int t = 2 * nPair4; t < nPairs; ++t) {
      acc += pair_energy<PACKED>(pairsFlat[2*t], pairsFlat[2*t + 1], pq, se,
                                 coords, charges, sigma, eps,
                                 Lx, Ly, Lz, iLx, iLy, iLz);
    }
  }
  float tot = block_sum256(acc);
  if (threadIdx.x == 0) partials[blockIdx.x] = tot;
}

__global__ void __launch_bounds__(256) reduce_partials_kernel(
    const float* __restrict__ partials, int n, float* __restrict__ out)
{
  float acc = 0.0f;
  for (int k = threadIdx.x; k < n; k += 256) acc += partials[k];
  float tot = block_sum256(acc);
  if (threadIdx.x == 0) out[0] = tot;
}

// ---------------------------------------------------------------------------
extern "C" void kernel_launch(void* const* d_in, const int* in_sizes, int n_in,
                              void* d_out, int out_size, void* d_ws, size_t ws_size,
                              hipStream_t stream)
{
  const float* coords  = (const float*)d_in[0];
  const float* box     = (const float*)d_in[1];
  const float* charges = (const float*)d_in[2];
  const float* sigma   = (const float*)d_in[3];
  const float* eps     = (const float*)d_in[4];
  const int*   pairs   = (const int*)d_in[5];

  const int n      = in_sizes[2];        // N particles (charges length)
  const int nPairs = in_sizes[5] / 2;    // pairs rows
  const int nPair4 = nPairs / 2;         // int4-packed pair count

  const int BLOCK = 256;
  const int GRID  = 2048;

  const size_t pqB = (size_t)n * sizeof(float4);
  const size_t seB = (size_t)n * sizeof(float2);
  const bool packed = (ws_size >= pqB + seB + (size_t)GRID * sizeof(float));

  float4* pq = (float4*)d_ws;
  float2* se = (float2*)((char*)d_ws + pqB);
  float*  partials = packed ? (float*)((char*)d_ws + pqB + seB)
                            : (float*)d_ws;

  if (packed) {
    pack_params_kernel<<<(n + BLOCK - 1) / BLOCK, BLOCK, 0, stream>>>(
        coords, charges, sigma, eps, pq, se, n);
    nb_energy_kernel<true><<<GRID, BLOCK, 0, stream>>>(
        (const int4*)pairs, nPair4, pairs, nPairs, pq, se,
        coords, charges, sigma, eps, box, partials);
  } else {
    nb_energy_kernel<false><<<GRID, BLOCK, 0, stream>>>(
        (const int4*)pairs, nPair4, pairs, nPairs, nullptr, nullptr,
        coords, charges, sigma, eps, box, partials);
  }
  reduce_partials_kernel<<<1, BLOCK, 0, stream>>>(partials, GRID, (float*)d_out);
}